// Enhance_model_73083163508823
// MI455X (gfx1250) — compile-verified
//
#include <hip/hip_runtime.h>
#include <math.h>

// ---- problem constants (match reference) ----
#define BB      16
#define TDIM    12
#define BT      192      // B*T
#define NN      512
#define DD      64
#define EDS     32
#define HS      16
#define HT      32
#define TTK     192      // TT = HS*T
#define ED      32
#define NROUTE  3

typedef __attribute__((ext_vector_type(2))) float v2f;
typedef __attribute__((ext_vector_type(8))) float v8f;

// CDNA5 fp32 WMMA: D(16x16,f32) = A(16x4,f32) x B(4x16,f32) + C
// args: (neg_a, A, neg_b, B, c_mod, C, reuse_a, reuse_b)
static __device__ __forceinline__ v8f wmma4(v2f a, v2f b, v8f c) {
  return __builtin_amdgcn_wmma_f32_16x16x4_f32(false, a, false, b, (short)0, c,
                                               false, false);
}

static __device__ __forceinline__ float lrelu(float x) {
  return x >= 0.f ? x : 0.01f * x;
}
// squash(x) = (sq/(1+sq)) * x / (sqrt(sq)+1e-8), sq = sum(x^2)
static __device__ __forceinline__ float squash_scale(float sq) {
  return (sq / (1.f + sq)) / (sqrtf(sq) + 1e-8f);
}

// =====================================================================
// K1: Pc = squash(x @ W_p + b_p), rows = BT*N = 98304, K=N=64
// one wave per 16-row tile, 4 col tiles of 16, WMMA f32 16x16x4
// =====================================================================
__global__ void __launch_bounds__(32) k_pc(const float* __restrict__ x,
                                           const float* __restrict__ Wp,
                                           const float* __restrict__ bp,
                                           float* __restrict__ Pc) {
  const int lane = threadIdx.x & 31;
  const int half = lane >> 4, l16 = lane & 15;
  const long rowBase = (long)blockIdx.x * 16;
  const float* arow = x + (rowBase + l16) * DD;
  v8f acc[4] = {};
  for (int k0 = 0; k0 < DD; k0 += 4) {
    v2f a;
    a.x = arow[k0 + 2 * half];
    a.y = arow[k0 + 2 * half + 1];
#pragma unroll
    for (int ct = 0; ct < 4; ++ct) {
      v2f b;
      b.x = Wp[(k0 + 2 * half) * DD + ct * 16 + l16];
      b.y = Wp[(k0 + 2 * half + 1) * DD + ct * 16 + l16];
      acc[ct] = wmma4(a, b, acc[ct]);
    }
  }
#pragma unroll
  for (int ct = 0; ct < 4; ++ct) {
    float bias = bp[ct * 16 + l16];
#pragma unroll
    for (int j = 0; j < 8; ++j) acc[ct][j] += bias;
  }
  // per-row squash: row of VGPR j = j + 8*half, 64 cols spread over l16 x 4 tiles
#pragma unroll
  for (int j = 0; j < 8; ++j) {
    float sq = 0.f;
#pragma unroll
    for (int ct = 0; ct < 4; ++ct) sq += acc[ct][j] * acc[ct][j];
    sq += __shfl_xor(sq, 1, 16);
    sq += __shfl_xor(sq, 2, 16);
    sq += __shfl_xor(sq, 4, 16);
    sq += __shfl_xor(sq, 8, 16);
    float sc = squash_scale(sq);
    float* orow = Pc + (rowBase + j + 8 * half) * DD;
#pragma unroll
    for (int ct = 0; ct < 4; ++ct) orow[ct * 16 + l16] = acc[ct][j] * sc;
  }
}

// =====================================================================
// K2: dadj[bt,h,n] = sum_d teb[bt,d]*adj[d,h,n]
// =====================================================================
__global__ void k_dadj(const float* __restrict__ teb,
                       const float* __restrict__ adj,
                       float* __restrict__ dadj) {
  int idx = blockIdx.x * blockDim.x + threadIdx.x;
  if (idx >= BT * HS * NN) return;
  int n = idx % NN;
  int h = (idx / NN) % HS;
  int bt = idx / (NN * HS);
  const float* t = teb + bt * EDS;
  float s = 0.f;
#pragma unroll
  for (int d = 0; d < EDS; ++d) s += t[d] * adj[(d * HS + h) * NN + n];
  dadj[idx] = s;
}

// =====================================================================
// K3: smd = softmax(dadj) over the HS axis  (per (bt,n) column of 16)
// =====================================================================
__global__ void k_smdadj(const float* __restrict__ dadj,
                         float* __restrict__ smd) {
  int idx = blockIdx.x * blockDim.x + threadIdx.x;  // BT*N
  if (idx >= BT * NN) return;
  int n = idx % NN, bt = idx / NN;
  const float* p = dadj + (long)bt * HS * NN + n;
  float m = -1e30f;
#pragma unroll
  for (int h = 0; h < HS; ++h) m = fmaxf(m, p[h * NN]);
  float e[HS], z = 0.f;
#pragma unroll
  for (int h = 0; h < HS; ++h) { e[h] = __expf(p[h * NN] - m); z += e[h]; }
  float inv = 1.f / z;
  float* q = smd + (long)bt * HS * NN + n;
#pragma unroll
  for (int h = 0; h < HS; ++h) q[h * NN] = e[h] * inv;
}

// =====================================================================
// K4: sq1[bt,h,:] = squash_d( sum_n smd[bt,h,n]*Pc[bt,n,:] )
// one 64-thread block per (bt,h)
// =====================================================================
__global__ void __launch_bounds__(64) k_test1(const float* __restrict__ smd,
                                              const float* __restrict__ Pc,
                                              float* __restrict__ sq1) {
  int g = blockIdx.x;  // BT*HS
  int h = g % HS, bt = g / HS;
  int d = threadIdx.x;
  const float* c = smd + (long)(bt * HS + h) * NN;
  const float* p = Pc + (long)bt * NN * DD + d;
  float s = 0.f;
  for (int n = 0; n < NN; ++n) s += c[n] * p[n * DD];
  float sq = s * s;
  for (int m = 1; m < 32; m <<= 1) sq += __shfl_xor(sq, m, 32);
  __shared__ float tmp[2];
  if ((threadIdx.x & 31) == 0) tmp[threadIdx.x >> 5] = sq;
  __syncthreads();
  sq1[(long)g * DD + d] = s * squash_scale(tmp[0] + tmp[1]);
}

// =====================================================================
// K5: fused dynamic routing, one 256-thread workgroup (8 waves) per (b,t).
// u_hat never materialized:  s = sq1 (.) (c @ Pc).
// c@Pc (M=16,K=512,Nc=64)  : waves 0..3, one 16x16 col tile each (WMMA)
// v@Pc^T (M=16,K=64,Nc=512): 8 waves x 4 col tiles (WMMA)
// b_log / c live in L2 (global scratch), s/v/sq1 tiles in LDS.
// =====================================================================
__global__ void __launch_bounds__(256) k_route(const float* __restrict__ Pc,
                                               const float* __restrict__ dadj,
                                               const float* __restrict__ sq1,
                                               float* __restrict__ blog,
                                               float* __restrict__ cout,
                                               float* __restrict__ sout) {
  const int bt = blockIdx.x;
  const int tid = threadIdx.x;
  const int wave = tid >> 5, lane = tid & 31;
  const int half = lane >> 4, l16 = lane & 15;
  const float* Pcb = Pc + (long)bt * NN * DD;
  const float* dab = dadj + (long)bt * HS * NN;
  const float* sqb = sq1 + (long)bt * HS * DD;
  float* blb = blog + (long)bt * HS * NN;
  float* cb = cout + (long)bt * HS * NN;

  __shared__ float s_lds[HS * DD];   // s, then v (4KB)
  __shared__ float sq_lds[HS * DD];  // squash(test1) tile (4KB)
  __shared__ float red[HS];

  for (int i = tid; i < HS * DD; i += 256) sq_lds[i] = sqb[i];
  for (int i = tid; i < HS * NN; i += 256) blb[i] = 0.f;  // b_log = 0
  __threadfence_block();
  __syncthreads();

  for (int it = 0; it < NROUTE; ++it) {
    // c = softmax_h(b_log)
    for (int n = tid; n < NN; n += 256) {
      float m = -1e30f;
#pragma unroll
      for (int h = 0; h < HS; ++h) m = fmaxf(m, blb[h * NN + n]);
      float e[HS], z = 0.f;
#pragma unroll
      for (int h = 0; h < HS; ++h) { e[h] = __expf(blb[h * NN + n] - m); z += e[h]; }
      float inv = 1.f / z;
#pragma unroll
      for (int h = 0; h < HS; ++h) cb[h * NN + n] = e[h] * inv;
    }
    __threadfence_block();
    __syncthreads();

    // s = sq1 (.) (c @ Pc)   -- WMMA, waves 0..3 (EXEC all-ones per wave)
    if (wave < 4) {
      const int ct = wave;
      v8f acc = {};
      for (int k0 = 0; k0 < NN; k0 += 4) {
        v2f a, b;
        a.x = cb[l16 * NN + k0 + 2 * half];
        a.y = cb[l16 * NN + k0 + 2 * half + 1];
        b.x = Pcb[(k0 + 2 * half) * DD + ct * 16 + l16];
        b.y = Pcb[(k0 + 2 * half + 1) * DD + ct * 16 + l16];
        acc = wmma4(a, b, acc);
      }
#pragma unroll
      for (int j = 0; j < 8; ++j) {
        int idx = (j + 8 * half) * DD + ct * 16 + l16;
        s_lds[idx] = sq_lds[idx] * acc[j];
      }
    }
    __syncthreads();

    // v = squash_d(s) in LDS
    if (tid < HS) {
      float sq = 0.f;
#pragma unroll
      for (int d = 0; d < DD; ++d) { float v = s_lds[tid * DD + d]; sq += v * v; }
      red[tid] = squash_scale(sq);
    }
    __syncthreads();
    for (int i = tid; i < HS * DD; i += 256) s_lds[i] *= red[i >> 6];
    __syncthreads();

    // b_log += v @ Pc^T   -- WMMA, 8 waves x 4 col tiles of n
#pragma unroll
    for (int cc = 0; cc < 4; ++cc) {
      const int ct = wave * 4 + cc;
      v8f acc = {};
#pragma unroll
      for (int k0 = 0; k0 < DD; k0 += 4) {
        v2f a, b;
        a.x = s_lds[l16 * DD + k0 + 2 * half];
        a.y = s_lds[l16 * DD + k0 + 2 * half + 1];
        b.x = Pcb[(ct * 16 + l16) * DD + k0 + 2 * half];      // B[d][n]=Pc[n][d]
        b.y = Pcb[(ct * 16 + l16) * DD + k0 + 2 * half + 1];
        acc = wmma4(a, b, acc);
      }
#pragma unroll
      for (int j = 0; j < 8; ++j)
        blb[(j + 8 * half) * NN + ct * 16 + l16] += acc[j];
    }
    __threadfence_block();
    __syncthreads();
  }

  // final c = softmax_h(b_log + dadj)
  for (int n = tid; n < NN; n += 256) {
    float m = -1e30f;
#pragma unroll
    for (int h = 0; h < HS; ++h) m = fmaxf(m, blb[h * NN + n] + dab[h * NN + n]);
    float e[HS], z = 0.f;
#pragma unroll
    for (int h = 0; h < HS; ++h) {
      e[h] = __expf(blb[h * NN + n] + dab[h * NN + n] - m);
      z += e[h];
    }
    float inv = 1.f / z;
#pragma unroll
    for (int h = 0; h < HS; ++h) cb[h * NN + n] = e[h] * inv;
  }
  __threadfence_block();
  __syncthreads();

  // s_final = c @ Pc  -> sout[bt]
  if (wave < 4) {
    const int ct = wave;
    v8f acc = {};
    for (int k0 = 0; k0 < NN; k0 += 4) {
      v2f a, b;
      a.x = cb[l16 * NN + k0 + 2 * half];
      a.y = cb[l16 * NN + k0 + 2 * half + 1];
      b.x = Pcb[(k0 + 2 * half) * DD + ct * 16 + l16];
      b.y = Pcb[(k0 + 2 * half + 1) * DD + ct * 16 + l16];
      acc = wmma4(a, b, acc);
    }
#pragma unroll
    for (int j = 0; j < 8; ++j)
      sout[(long)bt * HS * DD + (j + 8 * half) * DD + ct * 16 + l16] = acc[j];
  }
}

// =====================================================================
// K6: dyn[b,h,k] = sum_d time_eb[b,d]*t_adj[d,h,k]   (output #3)
// =====================================================================
__global__ void k_dyn(const float* __restrict__ time_eb,
                      const float* __restrict__ tadj,
                      float* __restrict__ dyn) {
  int idx = blockIdx.x * blockDim.x + threadIdx.x;
  if (idx >= BB * HT * TTK) return;
  int k = idx % TTK;
  int h = (idx / TTK) % HT;
  int b = idx / (TTK * HT);
  float s = 0.f;
#pragma unroll
  for (int d = 0; d < EDS; ++d)
    s += time_eb[b * EDS + d] * tadj[(d * HT + h) * TTK + k];
  dyn[idx] = s;
}

// =====================================================================
// K7: hyper_tem[b,h,d] = lrelu( sum_k dyn[b,h,k]*(s[b,k,d] + mask[k/HS]) )
// =====================================================================
__global__ void k_htem(const float* __restrict__ dyn,
                       const float* __restrict__ s, float* __restrict__ ht) {
  int idx = blockIdx.x * blockDim.x + threadIdx.x;
  if (idx >= BB * HT * DD) return;
  int d = idx % DD;
  int h = (idx / DD) % HT;
  int b = idx / (DD * HT);
  const float* dy = dyn + (b * HT + h) * TTK;
  const float* sb = s + (long)b * TTK * DD;
  float acc = 0.f;
  for (int k = 0; k < TTK; ++k) {
    float mask = (float)(k / HS + 1) * (1.0f / 12.0f);  // linspace(1,T,T)/12
    acc += dy[k] * (sb[k * DD + d] + mask);
  }
  ht[idx] = lrelu(acc);
}

// =====================================================================
// K8: ret[b,k,d] = lrelu( sum_h dyn[b,h,k]*ht[b,h,d] )
// =====================================================================
__global__ void k_ret(const float* __restrict__ dyn,
                      const float* __restrict__ ht, float* __restrict__ ret) {
  int idx = blockIdx.x * blockDim.x + threadIdx.x;
  if (idx >= BB * TTK * DD) return;
  int d = idx % DD;
  int k = (idx / DD) % TTK;
  int b = idx / (DD * TTK);
  float acc = 0.f;
#pragma unroll
  for (int h = 0; h < HT; ++h)
    acc += dyn[(b * HT + h) * TTK + k] * ht[(b * HT + h) * DD + d];
  ret[idx] = lrelu(acc);
}

// =====================================================================
// K9: v[g,:] = squash_d(ret[g,:] + s[g,:]),  g over BT*HS (same flattening)
// =====================================================================
__global__ void __launch_bounds__(64) k_vfin(const float* __restrict__ ret,
                                             const float* __restrict__ s,
                                             float* __restrict__ v) {
  int g = blockIdx.x;
  int d = threadIdx.x;
  float val = ret[(long)g * DD + d] + s[(long)g * DD + d];
  float sq = val * val;
  for (int m = 1; m < 32; m <<= 1) sq += __shfl_xor(sq, m, 32);
  __shared__ float tmp[2];
  if ((threadIdx.x & 31) == 0) tmp[threadIdx.x >> 5] = sq;
  __syncthreads();
  v[(long)g * DD + d] = val * squash_scale(tmp[0] + tmp[1]);
}

// =====================================================================
// K10: recon[bt,n,d] = sum_h c[bt,h,n]*v[bt,h,d]   (K=16, bandwidth bound)
// =====================================================================
__global__ void k_recon(const float* __restrict__ c,
                        const float* __restrict__ v,
                        float* __restrict__ recon) {
  long idx = (long)blockIdx.x * blockDim.x + threadIdx.x;
  if (idx >= (long)BT * NN * DD) return;
  int d = (int)(idx % DD);
  int n = (int)((idx / DD) % NN);
  int bt = (int)(idx / ((long)DD * NN));
  const float* cbp = c + (long)bt * HS * NN;
  const float* vb = v + (long)bt * HS * DD;
  float acc = 0.f;
#pragma unroll
  for (int h = 0; h < HS; ++h) acc += cbp[h * NN + n] * vb[h * DD + d];
  recon[idx] = acc;
}

// =====================================================================
// K11: W_spatial[n, io] = sum_d ne[n,d]*weights_spa[d, io]
// GEMM M=512, K=32, Ncols=4096 -- WMMA, 32 row tiles x 64 col groups
// =====================================================================
__global__ void __launch_bounds__(32) k_wsp(const float* __restrict__ ne,
                                            const float* __restrict__ wspa,
                                            float* __restrict__ Wsp) {
  int rb = (blockIdx.x >> 6) * 16;
  int cg = (blockIdx.x & 63) * 64;
  int lane = threadIdx.x & 31, half = lane >> 4, l16 = lane & 15;
  const float* ar = ne + (rb + l16) * ED;
  v8f acc[4] = {};
  for (int k0 = 0; k0 < ED; k0 += 4) {
    v2f a;
    a.x = ar[k0 + 2 * half];
    a.y = ar[k0 + 2 * half + 1];
#pragma unroll
    for (int ct = 0; ct < 4; ++ct) {
      v2f b;
      b.x = wspa[(k0 + 2 * half) * (DD * DD) + cg + ct * 16 + l16];
      b.y = wspa[(k0 + 2 * half + 1) * (DD * DD) + cg + ct * 16 + l16];
      acc[ct] = wmma4(a, b, acc[ct]);
    }
  }
#pragma unroll
  for (int j = 0; j < 8; ++j) {
    long row = rb + j + 8 * half;
#pragma unroll
    for (int ct = 0; ct < 4; ++ct)
      Wsp[row * (DD * DD) + cg + ct * 16 + l16] = acc[ct][j];
  }
}

// =====================================================================
// K12: b_spatial[n,d] = sum_e ne[n,e]*bias_spa[e,d]
// =====================================================================
__global__ void k_bsp(const float* __restrict__ ne,
                      const float* __restrict__ bspa,
                      float* __restrict__ bsp) {
  int idx = blockIdx.x * blockDim.x + threadIdx.x;
  if (idx >= NN * DD) return;
  int d = idx % DD, n = idx / DD;
  float acc = 0.f;
#pragma unroll
  for (int e = 0; e < ED; ++e) acc += ne[n * ED + e] * bspa[e * DD + d];
  bsp[idx] = acc;
}

// =====================================================================
// K13: out = lrelu(recon[.,n,:] @ W_spatial[n] + b_spatial[n] + x)
// per-n GEMM [192x64]x[64x64] -- WMMA, grid = N * (BT/16)
// =====================================================================
__global__ void __launch_bounds__(32) k_out(const float* __restrict__ recon,
                                            const float* __restrict__ Wsp,
                                            const float* __restrict__ bsp,
                                            const float* __restrict__ x,
                                            float* __restrict__ out) {
  int n = blockIdx.x % NN;
  int rb = (blockIdx.x / NN) * 16;  // bt tile
  int lane = threadIdx.x & 31, half = lane >> 4, l16 = lane & 15;
  const float* Wn = Wsp + (long)n * DD * DD;
  const float* ar = recon + ((long)(rb + l16) * NN + n) * DD;
  v8f acc[4] = {};
  for (int k0 = 0; k0 < DD; k0 += 4) {
    v2f a;
    a.x = ar[k0 + 2 * half];
    a.y = ar[k0 + 2 * half + 1];
#pragma unroll
    for (int ct = 0; ct < 4; ++ct) {
      v2f b;
      b.x = Wn[(k0 + 2 * half) * DD + ct * 16 + l16];
      b.y = Wn[(k0 + 2 * half + 1) * DD + ct * 16 + l16];
      acc[ct] = wmma4(a, b, acc[ct]);
    }
  }
#pragma unroll
  for (int j = 0; j < 8; ++j) {
    long base = ((long)(rb + j + 8 * half) * NN + n) * DD;
#pragma unroll
    for (int ct = 0; ct < 4; ++ct) {
      int col = ct * 16 + l16;
      float vv = acc[ct][j] + bsp[n * DD + col] + x[base + col];
      out[base + col] = lrelu(vv);
    }
  }
}

// =====================================================================
// host launcher
// =====================================================================
extern "C" void kernel_launch(void* const* d_in, const int* in_sizes, int n_in,
                              void* d_out, int out_size, void* d_ws,
                              size_t ws_size, hipStream_t stream) {
  (void)in_sizes; (void)n_in; (void)out_size; (void)ws_size;
  const float* x       = (const float*)d_in[0];
  const float* ne      = (const float*)d_in[1];
  const float* time_eb = (const float*)d_in[2];
  const float* teb     = (const float*)d_in[3];
  const float* Wp      = (const float*)d_in[4];
  const float* bp      = (const float*)d_in[5];
  const float* tadj    = (const float*)d_in[6];
  const float* adj     = (const float*)d_in[7];
  const float* wspa    = (const float*)d_in[8];
  const float* bspa    = (const float*)d_in[9];

  // outputs, concatenated flat: out1 [BT,N,D] | c [BT,HS,N,1] | dyn [B,HT,TT]
  float* out1 = (float*)d_out;
  float* cout = out1 + (long)BT * NN * DD;
  float* dyn  = cout + (long)BT * HS * NN;

  // workspace carve-up (~81 MB of fp32 scratch)
  float* ws   = (float*)d_ws;
  float* Pc   = ws; ws += (long)BT * NN * DD;   // 6.29M
  float* dadj = ws; ws += (long)BT * HS * NN;   // 1.57M
  float* smd  = ws; ws += (long)BT * HS * NN;   // 1.57M
  float* blog = ws; ws += (long)BT * HS * NN;   // 1.57M
  float* sq1  = ws; ws += (long)BT * HS * DD;   // 0.20M
  float* sbuf = ws; ws += (long)BT * HS * DD;   // 0.20M
  float* htem = ws; ws += (long)BB * HT * DD;   // 0.03M
  float* ret  = ws; ws += (long)BB * TTK * DD;  // 0.20M
  float* vbuf = ws; ws += (long)BT * HS * DD;   // 0.20M
  float* recon= ws; ws += (long)BT * NN * DD;   // 6.29M
  float* Wsp  = ws; ws += (long)NN * DD * DD;   // 2.10M
  float* bsp  = ws; ws += (long)NN * DD;        // 0.03M

  k_pc    <<<BT * NN / 16, 32, 0, stream>>>(x, Wp, bp, Pc);
  k_dadj  <<<(BT * HS * NN + 255) / 256, 256, 0, stream>>>(teb, adj, dadj);
  k_smdadj<<<(BT * NN + 255) / 256, 256, 0, stream>>>(dadj, smd);
  k_test1 <<<BT * HS, 64, 0, stream>>>(smd, Pc, sq1);
  k_route <<<BT, 256, 0, stream>>>(Pc, dadj, sq1, blog, cout, sbuf);
  k_dyn   <<<(BB * HT * TTK + 255) / 256, 256, 0, stream>>>(time_eb, tadj, dyn);
  k_htem  <<<(BB * HT * DD + 255) / 256, 256, 0, stream>>>(dyn, sbuf, htem);
  k_ret   <<<(BB * TTK * DD + 255) / 256, 256, 0, stream>>>(dyn, htem, ret);
  k_vfin  <<<BT * HS, 64, 0, stream>>>(ret, sbuf, vbuf);
  k_recon <<<(int)(((long)BT * NN * DD + 255) / 256), 256, 0, stream>>>(cout, vbuf, recon);
  k_wsp   <<<32 * 64, 32, 0, stream>>>(ne, wspa, Wsp);
  k_bsp   <<<(NN * DD + 255) / 256, 256, 0, stream>>>(ne, bspa, bsp);
  k_out   <<<NN * (BT / 16), 32, 0, stream>>>(recon, Wsp, bsp, x, out1);
}